// SpGQAFlashDecodeIris_12060268167434
// MI455X (gfx1250) — compile-verified
//
#include <hip/hip_runtime.h>

// dims from the reference
#define R_   4
#define B_   16
#define HQ_  32
#define HKV_ 2
#define G_   16
#define D_   128
#define S_   32
#define L_   2048
#define P_   32768

#define NEGBIG (-1e30f)

typedef float v2f __attribute__((ext_vector_type(2)));
typedef float v4f __attribute__((ext_vector_type(4)));
typedef float v8f __attribute__((ext_vector_type(8)));

// ---------------------------------------------------------------------------
// Stage 1: split-KV flash decode. One wave (32 lanes) per (b, kvh, rank, split).
// K/V tiles staged with async global->LDS copies (ASYNCcnt), double-buffered so
// tile t+1's gather overlaps tile t's WMMA + softmax work.
// ---------------------------------------------------------------------------
__global__ __launch_bounds__(32)
void gqa_decode_split_kernel(const float* __restrict__ q,
                             const float* __restrict__ k_cache,
                             const float* __restrict__ v_cache,
                             const int*   __restrict__ block_table,
                             const int*   __restrict__ kv_lens,
                             float*       __restrict__ ws_acc,
                             float2*      __restrict__ ws_ml)
{
    const int lane = threadIdx.x;
    const int b    = blockIdx.x / HKV_;
    const int kvh  = blockIdx.x % HKV_;
    const int r    = blockIdx.y;
    const int s    = blockIdx.z;
    const int sg   = r * S_ + s;            // global split id

    const int len   = kv_lens[r * B_ + b];
    const int per   = (len + S_ - 1) / S_;  // ceil(len / S)
    const int start = s * per;
    int npos = len - start;
    if (npos > per) npos = per;
    if (npos < 0)   npos = 0;
    const int ntiles = (npos + 15) >> 4;

    const int col  = lane & 15;             // N-index / A-row index
    const int half = lane >> 4;             // 0: lanes 0-15, 1: lanes 16-31
    const int koff = half * 2;              // A/B K-offset per 16x16x4 layout

    __shared__ float ldsK[2][16 * D_];      // 2 x 8 KB: double-buffered K tiles
    __shared__ float ldsV[2][16 * D_];      // 2 x 8 KB: double-buffered V tiles
    __shared__ float ldsP[16 * 16];         // 1 KB: P tile relayout C->A

    // Preload Q A-fragments: A is 16(G) x 4(Dchunk) f32 per WMMA.
    v2f qa[32];
    {
        const float* qrow = q + (size_t)(b * HQ_ + kvh * G_ + col) * D_;
        #pragma unroll
        for (int c = 0; c < 32; ++c)
            qa[c] = *(const v2f*)(qrow + 4 * c + koff);
    }

    const int* bt = block_table + ((size_t)r * B_ + b) * L_;
    const float scale = 0.088388347648318447f;   // D^-0.5

    v8f acc[8];
    #pragma unroll
    for (int j = 0; j < 8; ++j) acc[j] = (v8f){0,0,0,0,0,0,0,0};
    float mrow[8], lrow[8];
    #pragma unroll
    for (int vi = 0; vi < 8; ++vi) { mrow[vi] = NEGBIG; lrow[vi] = 0.f; }

    // Stage one 16-token K/V tile into LDS buffer `buf` with async copies.
    // Each valid token row: 32 lanes x 16B = 512B per matrix, 2 async ops/row.
    auto stage_tile = [&](int t, int buf) {
        #pragma unroll 4
        for (int i = 0; i < 16; ++i) {
            const int pcol = t * 16 + i;     // uniform across wave
            float* lkp = &ldsK[buf][i * D_ + lane * 4];
            float* lvp = &ldsV[buf][i * D_ + lane * 4];
            if (pcol < npos) {
                const int page = bt[start + pcol];
                const size_t rowoff = (((size_t)r * P_ + page) * HKV_ + kvh) * (size_t)D_;
                const float* kp = k_cache + rowoff + lane * 4;
                const float* vp = v_cache + rowoff + lane * 4;
                const unsigned lk = (unsigned)(size_t)lkp;
                const unsigned lv = (unsigned)(size_t)lvp;
                asm volatile("global_load_async_to_lds_b128 %0, %1, off"
                             :: "v"(lk), "v"(kp) : "memory");
                asm volatile("global_load_async_to_lds_b128 %0, %1, off"
                             :: "v"(lv), "v"(vp) : "memory");
            } else {
                const v4f z = (v4f){0,0,0,0};
                *(v4f*)lkp = z;              // plain ds_store; same-wave LDS is in-order
                *(v4f*)lvp = z;
            }
        }
    };

    if (ntiles > 0) stage_tile(0, 0);

    for (int t = 0; t < ntiles; ++t) {
        const int buf = t & 1;
        const bool have_next = (t + 1) < ntiles;
        if (have_next) stage_tile(t + 1, buf ^ 1);

        // Async loads complete in order: waiting "<= 32 outstanding" after issuing
        // tile t+1's 32 ops guarantees all of tile t's ops have landed in LDS.
        if (have_next) asm volatile("s_wait_asynccnt 0x20" ::: "memory");
        else           asm volatile("s_wait_asynccnt 0x0"  ::: "memory");
        __syncthreads();

        // ---- QK^T: scores[16 G x 16 pos] via 32 chained V_WMMA_F32_16X16X4_F32 ----
        v8f sc = (v8f){0,0,0,0,0,0,0,0};
        #pragma unroll
        for (int c = 0; c < 32; ++c) {
            v2f bf = *(const v2f*)(&ldsK[buf][col * D_ + 4 * c + koff]);
            sc = __builtin_amdgcn_wmma_f32_16x16x4_f32(false, qa[c], false, bf,
                                                       (short)0, sc, false, false);
        }

        // ---- online softmax; rows live in 16-lane halves (C layout: M = vi + 8*half) ----
        const bool validc = (t * 16 + col) < npos;
        #pragma unroll
        for (int vi = 0; vi < 8; ++vi) {
            float sv = validc ? sc[vi] * scale : NEGBIG;
            float tm = sv;
            tm = fmaxf(tm, __shfl_xor(tm, 1, 32));
            tm = fmaxf(tm, __shfl_xor(tm, 2, 32));
            tm = fmaxf(tm, __shfl_xor(tm, 4, 32));
            tm = fmaxf(tm, __shfl_xor(tm, 8, 32));
            const float mn    = fmaxf(mrow[vi], tm);
            const float alpha = __expf(mrow[vi] - mn);
            float pv = validc ? __expf(sv - mn) : 0.f;   // masked cols contribute 0
            float rs = pv;
            rs += __shfl_xor(rs, 1, 32);
            rs += __shfl_xor(rs, 2, 32);
            rs += __shfl_xor(rs, 4, 32);
            rs += __shfl_xor(rs, 8, 32);
            lrow[vi] = lrow[vi] * alpha + rs;
            mrow[vi] = mn;
            #pragma unroll
            for (int j = 0; j < 8; ++j) acc[j][vi] *= alpha;
            ldsP[(vi + 8 * half) * 16 + col] = pv;       // P in [g][pos] for relayout
        }
        __syncthreads();

        // ---- PV: acc[16 G x 128 D] += P[16 x 16] * V[16 x 128] ----
        #pragma unroll
        for (int cc = 0; cc < 4; ++cc) {
            v2f af;      // A = P: row g = lane&15, K(pos) = cc*4 + koff(+1)
            af.x = ldsP[col * 16 + cc * 4 + koff];
            af.y = ldsP[col * 16 + cc * 4 + koff + 1];
            #pragma unroll
            for (int j = 0; j < 8; ++j) {
                v2f bv;  // B = V: col d = j*16 + (lane&15), K(pos) rows of V tile
                bv.x = ldsV[buf][(cc * 4 + koff)     * D_ + j * 16 + col];
                bv.y = ldsV[buf][(cc * 4 + koff + 1) * D_ + j * 16 + col];
                acc[j] = __builtin_amdgcn_wmma_f32_16x16x4_f32(false, af, false, bv,
                                                               (short)0, acc[j], false, false);
            }
        }
        __syncthreads();
    }

    // ---- store unnormalized partial acc + (m, l) ----
    float* accp = ws_acc + (((size_t)sg * B_ + b) * HKV_ + kvh) * (size_t)(G_ * D_);
    #pragma unroll
    for (int vi = 0; vi < 8; ++vi) {
        const int Mr = vi + 8 * half;
        #pragma unroll
        for (int j = 0; j < 8; ++j)
            accp[Mr * D_ + j * 16 + col] = acc[j][vi];
    }
    if (col == 0) {  // lanes 0 and 16 each own 8 rows of (m, l)
        float2* mlp = ws_ml + ((size_t)sg * B_ + b) * HQ_ + kvh * G_;
        #pragma unroll
        for (int vi = 0; vi < 8; ++vi)
            mlp[vi + 8 * half] = make_float2(mrow[vi], lrow[vi]);
    }
}

// ---------------------------------------------------------------------------
// Stage 2: exact global LSE-weighted combine across all R*S partials.
// ---------------------------------------------------------------------------
__global__ __launch_bounds__(128)
void gqa_decode_combine_kernel(const float*  __restrict__ ws_acc,
                               const float2* __restrict__ ws_ml,
                               float*        __restrict__ out)
{
    const int d   = threadIdx.x;
    const int h   = blockIdx.x % HQ_;
    const int b   = blockIdx.x / HQ_;
    const int kvh = h / G_;
    const int g   = h % G_;

    float M = NEGBIG;
    for (int sg = 0; sg < R_ * S_; ++sg) {
        float2 t = ws_ml[((size_t)sg * B_ + b) * HQ_ + h];
        M = fmaxf(M, t.x);
    }
    float num = 0.f, den = 0.f;
    for (int sg = 0; sg < R_ * S_; ++sg) {
        float2 t = ws_ml[((size_t)sg * B_ + b) * HQ_ + h];
        const float w = __expf(t.x - M);
        den += w * t.y;
        num += w * ws_acc[((((size_t)sg * B_ + b) * HKV_ + kvh) * G_ + g) * (size_t)D_ + d];
    }
    out[((size_t)b * HQ_ + h) * D_ + d] = num / den;
}

// ---------------------------------------------------------------------------
extern "C" void kernel_launch(void* const* d_in, const int* in_sizes, int n_in,
                              void* d_out, int out_size, void* d_ws, size_t ws_size,
                              hipStream_t stream)
{
    const float* q           = (const float*)d_in[0];
    const float* k_cache     = (const float*)d_in[1];
    const float* v_cache     = (const float*)d_in[2];
    const int*   block_table = (const int*)d_in[3];
    const int*   kv_lens     = (const int*)d_in[4];

    const size_t accElems = (size_t)R_ * S_ * B_ * HKV_ * G_ * D_;  // 8.4M floats (32 MB)
    float*  ws_acc = (float*)d_ws;
    float2* ws_ml  = (float2*)((char*)d_ws + accElems * sizeof(float));

    dim3 grid1(B_ * HKV_, R_, S_);           // 4096 one-wave workgroups
    gqa_decode_split_kernel<<<grid1, dim3(32, 1, 1), 0, stream>>>(
        q, k_cache, v_cache, block_table, kv_lens, ws_acc, ws_ml);

    gqa_decode_combine_kernel<<<dim3(B_ * HQ_, 1, 1), dim3(128, 1, 1), 0, stream>>>(
        ws_acc, ws_ml, (float*)d_out);
}